// ConvictionPlanner_4131758539381
// MI455X (gfx1250) — compile-verified
//
#include <hip/hip_runtime.h>

#define A_N 5
#define D_N 10000
#define Z_N 512
#define B_N 256
#define KNN 50
#define EPS_K 1e-3f
#define DTILES 625            // D_N / 16
#define M_MLP 1536            // 256 pred rows + 1280 nstep rows

typedef __attribute__((ext_vector_type(2))) float v2f;
typedef __attribute__((ext_vector_type(8))) float v8f;

// ---------------------------------------------------------------------------
// keys_sq[a*D + d] = sum_z mem_keys[a][d][z]^2       (grid = A*D blocks, 128 thr)
// ---------------------------------------------------------------------------
__global__ void __launch_bounds__(128) cp_keys_sq(const float* __restrict__ mem_keys,
                                                  float* __restrict__ keys_sq) {
    __shared__ float red[128];
    const int row = blockIdx.x;
    const float* p = mem_keys + (size_t)row * Z_N;
    float s = 0.f;
    for (int k = threadIdx.x; k < Z_N; k += 128) { float v = p[k]; s += v * v; }
    red[threadIdx.x] = s; __syncthreads();
    for (int off = 64; off > 0; off >>= 1) {
        if (threadIdx.x < off) red[threadIdx.x] += red[threadIdx.x + off];
        __syncthreads();
    }
    if (threadIdx.x == 0) keys_sq[row] = red[0];
}

// ---------------------------------------------------------------------------
// Deterministic per-action sample compaction for step t (single thread).
// ---------------------------------------------------------------------------
__global__ void cp_prep_lists(const int* __restrict__ actions, int t,
                              int* __restrict__ lists, int* __restrict__ counts) {
    if (threadIdx.x == 0 && blockIdx.x == 0) {
        int cnt[A_N];
        for (int a = 0; a < A_N; ++a) cnt[a] = 0;
        for (int b = 0; b < B_N; ++b) {
            int a = actions[b * A_N + t];
            lists[a * B_N + cnt[a]] = b;
            cnt[a]++;
        }
        for (int a = 0; a < A_N; ++a) counts[a] = cnt[a];
    }
}

// ---------------------------------------------------------------------------
// qq[b] = dot(q_b, q_b)       (grid = B blocks, 128 thr)
// ---------------------------------------------------------------------------
__global__ void __launch_bounds__(128) cp_qq(const float* __restrict__ qbase, int qstride,
                                             float* __restrict__ qq) {
    __shared__ float red[128];
    const int b = blockIdx.x;
    const float* p = qbase + (size_t)b * qstride;
    float s = 0.f;
    for (int k = threadIdx.x; k < Z_N; k += 128) { float v = p[k]; s += v * v; }
    red[threadIdx.x] = s; __syncthreads();
    for (int off = 64; off > 0; off >>= 1) {
        if (threadIdx.x < off) red[threadIdx.x] += red[threadIdx.x + off];
        __syncthreads();
    }
    if (threadIdx.x == 0) qq[b] = red[0];
}

// ---------------------------------------------------------------------------
// dist[b][d] = keys_sq[a][d] - 2*(K_a @ q_b)[d] + qq[b]  via v_wmma_f32_16x16x4_f32.
//
// grid: (625 d-tiles, 4 sample-tile groups, A actions); block = 128 (4 waves).
// The block's 16x512 KEY tile (the HBM-heavy operand) is staged ONCE in LDS and
// shared by all 4 waves; each wave handles a different 16-sample tile, pulling
// its per-lane query float2 from global (queries are 512 KB total -> L2-hot).
// Net: mem_keys streamed from HBM exactly once per step.
// ---------------------------------------------------------------------------
__global__ void __launch_bounds__(128)
cp_dist_wmma(const float* __restrict__ mem_keys, const float* __restrict__ keys_sq,
             const float* __restrict__ qbase, int qstride,
             const float* __restrict__ qq,
             const int* __restrict__ lists, const int* __restrict__ counts,
             float* __restrict__ dist) {
    __shared__ __align__(16) float As[16 * Z_N];   // 32 KB key tile
    __shared__ int   scol_s[64];
    __shared__ float sqq_s[64];

    const int a  = blockIdx.z;
    const int na = counts[a];
    if ((int)blockIdx.y * 64 >= na) return;        // block-uniform

    const int tid = threadIdx.x;
    const int d0  = blockIdx.x * 16;

    if (tid < 64) {                                // sample ids for this block's 4 tiles
        int s = blockIdx.y * 64 + tid;
        int c = (s < na) ? s : blockIdx.y * 64;    // clamp pad slots to a valid sample
        int smp = lists[a * B_N + c];
        scol_s[tid] = smp;
        sqq_s[tid]  = qq[smp];
    }
    // cooperative float4 load of the key tile into LDS
    const float* Ag = mem_keys + ((size_t)a * D_N + d0) * Z_N;
    for (int v = tid; v < (16 * Z_N) / 4; v += 128) {
        ((float4*)As)[v] = ((const float4*)Ag)[v];
    }
    __syncthreads();

    const int wave  = tid >> 5;
    const int lane  = tid & 31;
    const int stile = blockIdx.y * 4 + wave;
    if (stile * 16 < na) {                         // wave-uniform
        const int nvalid = min(16, na - stile * 16);
        const int rc  = lane & 15;                 // A: M row (d); B: N col (sample)
        const int kk2 = (lane >> 4) * 2;           // K sub-offset {0,2}
        const int smp = scol_s[wave * 16 + rc];

        const float* Alds = &As[rc * Z_N + kk2];
        const float* Bg   = qbase + (size_t)smp * qstride + kk2;

        v8f acc = {};
#pragma unroll 8
        for (int k = 0; k < Z_N; k += 4) {
            v2f av = *(const v2f*)(Alds + k);      // ds_load_b64 (shared keys)
            v2f bv = *(const v2f*)(Bg + k);        // global_load_b64 (L2-hot queries)
            acc = __builtin_amdgcn_wmma_f32_16x16x4_f32(false, av, false, bv,
                                                        (short)0, acc, false, false);
        }

        const int mhi = (lane >> 4) * 8;           // C/D: lanes 16-31 hold M = r+8
        if (rc < nvalid) {
            const float qqv = sqq_s[wave * 16 + rc];
#pragma unroll
            for (int r = 0; r < 8; ++r) {
                int d = d0 + r + mhi;
                dist[(size_t)smp * D_N + d] = keys_sq[a * D_N + d] - 2.0f * acc[r] + qqv;
            }
        }
    }
}

// ---------------------------------------------------------------------------
// Top-50 (ascending dist, index tiebreak) + normalized inverse-distance weights.
// One block per sample; 10000 dists staged in LDS (40 KB), 50 argmin passes.
// ---------------------------------------------------------------------------
__global__ void __launch_bounds__(256)
cp_topk(const float* __restrict__ dist, int* __restrict__ topk_idx,
        float* __restrict__ topk_w) {
    __shared__ float sd[D_N];
    __shared__ float rmin[256];
    __shared__ int   ridx[256];
    __shared__ float wsum_s;
    const int b = blockIdx.x;
    const int tid = threadIdx.x;
    for (int i = tid; i < D_N; i += 256) sd[i] = dist[(size_t)b * D_N + i];
    if (tid == 0) wsum_s = 0.f;
    __syncthreads();
    for (int it = 0; it < KNN; ++it) {
        float best = 3.4e38f; int bi = D_N;
        for (int i = tid; i < D_N; i += 256) {
            float v = sd[i];
            if (v < best) { best = v; bi = i; }   // ascending i -> lowest index on tie
        }
        rmin[tid] = best; ridx[tid] = bi;
        __syncthreads();
        for (int off = 128; off > 0; off >>= 1) {
            if (tid < off) {
                float vo = rmin[tid + off]; int io = ridx[tid + off];
                if (vo < rmin[tid] || (vo == rmin[tid] && io < ridx[tid])) {
                    rmin[tid] = vo; ridx[tid] = io;
                }
            }
            __syncthreads();
        }
        if (tid == 0) {
            int idx = ridx[0];
            float w = 1.0f / (fmaxf(rmin[0], 0.f) + EPS_K);
            topk_idx[b * KNN + it] = idx;
            topk_w[b * KNN + it]  = w;
            wsum_s += w;
            sd[idx] = 3.4e38f;
        }
        __syncthreads();
    }
    const float inv = 1.0f / wsum_s;
    if (tid < KNN) topk_w[b * KNN + tid] *= inv;
}

// ---------------------------------------------------------------------------
// emb = sum_i w_i * V[a][idx_i]  ->  nstep_embs[b][t][:]
// ---------------------------------------------------------------------------
__global__ void __launch_bounds__(256)
cp_readout(const float* __restrict__ mem_vals, const int* __restrict__ actions, int t,
           const int* __restrict__ topk_idx, const float* __restrict__ topk_w,
           float* __restrict__ out_embs) {
    __shared__ float w_s[KNN];
    __shared__ int   i_s[KNN];
    const int b = blockIdx.x;
    const int tid = threadIdx.x;
    const int a = actions[b * A_N + t];
    if (tid < KNN) { w_s[tid] = topk_w[b * KNN + tid]; i_s[tid] = topk_idx[b * KNN + tid]; }
    __syncthreads();
    for (int z = tid; z < Z_N; z += 256) {
        float acc = 0.f;
#pragma unroll
        for (int i = 0; i < KNN; ++i)
            acc += w_s[i] * mem_vals[((size_t)a * D_N + i_s[i]) * Z_N + z];
        out_embs[((size_t)b * 5 + t) * Z_N + z] = acc;
    }
}

// ---------------------------------------------------------------------------
// X = concat(embeddings [256x512], nstep_embs [1280x512])
// ---------------------------------------------------------------------------
__global__ void cp_build_x(const float* __restrict__ embeddings,
                           const float* __restrict__ nstep_embs,
                           float* __restrict__ X) {
    const int i = blockIdx.x * 256 + threadIdx.x;          // 0 .. 1536*512-1
    X[i] = (i < B_N * Z_N) ? embeddings[i] : nstep_embs[i - B_N * Z_N];
}

// ---------------------------------------------------------------------------
// C[M,N] = elu?(X[M,K] @ W[K,N] + bias)   via v_wmma_f32_16x16x4_f32.
// grid: (M/16, N/16), one wave per tile.
// ---------------------------------------------------------------------------
__global__ void __launch_bounds__(32)
cp_mlp_wmma(const float* __restrict__ Xin, const float* __restrict__ W,
            const float* __restrict__ bias, float* __restrict__ Cout,
            int N, int K, int apply_elu) {
    const int m0 = blockIdx.x * 16;
    const int n0 = blockIdx.y * 16;
    const int lane = threadIdx.x;
    const int row  = lane & 15;
    const int kk2  = (lane >> 4) * 2;
    const float* Abase = Xin + (size_t)(m0 + row) * K + kk2;
    v8f acc = {};
    for (int k = 0; k < K; k += 4) {
        v2f av = *(const v2f*)(Abase + k);
        v2f bv;
        bv.x = W[(size_t)(k + kk2)     * N + n0 + row];
        bv.y = W[(size_t)(k + kk2 + 1) * N + n0 + row];
        acc = __builtin_amdgcn_wmma_f32_16x16x4_f32(false, av, false, bv,
                                                    (short)0, acc, false, false);
    }
    const int mhi = (lane >> 4) * 8;
#pragma unroll
    for (int r = 0; r < 8; ++r) {
        int m = m0 + r + mhi;
        int n = n0 + row;
        float v = acc[r] + bias[n];
        if (apply_elu) v = (v > 0.f) ? v : (__expf(v) - 1.0f);
        Cout[(size_t)m * N + n] = v;
    }
}

// ---------------------------------------------------------------------------
// Final 64 -> 1 layer; rows 0..255 -> pred_values, 256..1535 -> nstep_values.
// ---------------------------------------------------------------------------
__global__ void __launch_bounds__(256)
cp_mlp_final(const float* __restrict__ h3, const float* __restrict__ W4,
             const float* __restrict__ b4, float* __restrict__ out) {
    const int r = blockIdx.x * 256 + threadIdx.x;   // 0..1535
    float s = 0.f;
#pragma unroll
    for (int k = 0; k < 64; ++k) s += h3[r * 64 + k] * W4[k];
    out[r] = s + b4[0];
}

// ---------------------------------------------------------------------------
extern "C" void kernel_launch(void* const* d_in, const int* in_sizes, int n_in,
                              void* d_out, int out_size, void* d_ws, size_t ws_size,
                              hipStream_t stream) {
    const float* embeddings = (const float*)d_in[0];
    const int*   actions    = (const int*)d_in[1];
    const float* mem_keys   = (const float*)d_in[2];
    const float* mem_vals   = (const float*)d_in[3];
    const float* W1 = (const float*)d_in[4];  const float* b1 = (const float*)d_in[5];
    const float* W2 = (const float*)d_in[6];  const float* b2 = (const float*)d_in[7];
    const float* W3 = (const float*)d_in[8];  const float* b3 = (const float*)d_in[9];
    const float* W4 = (const float*)d_in[10]; const float* b4 = (const float*)d_in[11];

    float* out      = (float*)d_out;
    float* out_embs = out + B_N + B_N * 5;          // nstep_embs slice (offset 1536)

    // Workspace layout (floats); dist region is overlaid by X/h1/h2/h3 after step 5.
    float* ws       = (float*)d_ws;
    float* keys_sq  = ws;                           // 50000
    float* qq       = ws + 50000;                   // 256
    float* topk_w   = ws + 50256;                   // 12800
    int*   topk_idx = (int*)(ws + 63056);           // 12800
    int*   lists    = (int*)(ws + 75856);           // 1280
    int*   counts   = (int*)(ws + 77136);           // 8 (padded)
    float* dist     = ws + 77184;                   // 2,560,000
    float* X  = dist;                               // 786,432   (reuse)
    float* h1 = X  + M_MLP * 512;                   // 786,432
    float* h2 = h1 + M_MLP * 512;                   // 393,216
    float* h3 = h2 + M_MLP * 256;                   // 98,304

    cp_keys_sq<<<A_N * D_N, 128, 0, stream>>>(mem_keys, keys_sq);

    for (int t = 0; t < 5; ++t) {
        const float* qbase = (t == 0) ? embeddings : (out_embs + (t - 1) * Z_N);
        const int qstride  = (t == 0) ? Z_N : 5 * Z_N;

        cp_prep_lists<<<1, 32, 0, stream>>>(actions, t, lists, counts);
        cp_qq<<<B_N, 128, 0, stream>>>(qbase, qstride, qq);

        dim3 dg(DTILES, 4, A_N);                    // d-tile, sample-tile group, action
        cp_dist_wmma<<<dg, 128, 0, stream>>>(mem_keys, keys_sq, qbase, qstride,
                                             qq, lists, counts, dist);
        cp_topk<<<B_N, 256, 0, stream>>>(dist, topk_idx, topk_w);
        cp_readout<<<B_N, 256, 0, stream>>>(mem_vals, actions, t,
                                            topk_idx, topk_w, out_embs);
    }

    cp_build_x<<<(M_MLP * Z_N) / 256, 256, 0, stream>>>(embeddings, out_embs, X);
    cp_mlp_wmma<<<dim3(M_MLP / 16, 512 / 16), 32, 0, stream>>>(X,  W1, b1, h1, 512, 512, 1);
    cp_mlp_wmma<<<dim3(M_MLP / 16, 256 / 16), 32, 0, stream>>>(h1, W2, b2, h2, 256, 512, 1);
    cp_mlp_wmma<<<dim3(M_MLP / 16,  64 / 16), 32, 0, stream>>>(h2, W3, b3, h3,  64, 256, 1);
    cp_mlp_final<<<M_MLP / 256, 256, 0, stream>>>(h3, W4, b4, out);
}